// MultiHeadAttention_50216757624897
// MI455X (gfx1250) — compile-verified
//
#include <hip/hip_runtime.h>
#include <hip/hip_bf16.h>
#include <math.h>

// ---------------------------------------------------------------------------
// Decode-step MHA with KV cache on gfx1250 (MI455X).
//   BS=8, CTX=1, D_MODEL=1024, N_HEADS=16, HEAD_DIM=64, ROT=32, CACHE=4096
// Memory-bound: k_cache + v_cache = 256MB streamed once. Projections use
// fp32 WMMA (v_wmma_f32_16x16x4_f32); attention is a single-pass online
// softmax streaming kernel.
// ---------------------------------------------------------------------------

typedef float v2f __attribute__((ext_vector_type(2)));
typedef float v8f __attribute__((ext_vector_type(8)));

#define DM    1024
#define NH    16
#define HD    64
#define CL    4096
#define BS_   8

// One wave computes a 16(M) x 16(N) fp32 tile of  out = A(8x1024) @ W^T + b.
// A rows 8..15 are zero (M padded to 16).  W is row-major (N x K), so
// B[k][n] = W[n0+n][k] -> per-lane layout is symmetric with A.
__device__ __forceinline__ void tile16_gemm_wmma(
    const float* __restrict__ A,    // 8 x 1024, row-major
    const float* __restrict__ W,    // 1024 x 1024, row-major
    const float* __restrict__ bias, // 1024
    float* __restrict__ out,        // 8 x 1024
    int n0, int lane)
{
  const int   half  = lane >> 4;          // 0: K+0/K+1, 1: K+2/K+3
  const int   lm    = lane & 15;
  const float amask = (lm < 8) ? 1.0f : 0.0f;
  const int   arow  = lm & 7;             // always-valid address, masked value
  const int   wrow  = n0 + lm;

  v8f c = {0.f,0.f,0.f,0.f,0.f,0.f,0.f,0.f};

  #pragma unroll 4
  for (int k = 0; k < DM; k += 4) {
    const int ko = k + 2 * half;
    v2f a, b;
    const float2 av = *(const float2*)(A + arow * DM + ko);
    a.x = amask * av.x;
    a.y = amask * av.y;
    const float2 wv = *(const float2*)(W + wrow * DM + ko);
    b.x = wv.x;
    b.y = wv.y;
    // D = A(16x4) * B(4x16) + C ; full fp32 matrix core path
    c = __builtin_amdgcn_wmma_f32_16x16x4_f32(
        /*neg_a=*/false, a, /*neg_b=*/false, b,
        /*c_mod=*/(short)0, c, /*reuse_a=*/false, /*reuse_b=*/false);
  }

  // D layout: VGPR r, lanes 0..15 -> D[M=r][N=lane]; lanes 16..31 hold the
  // padded rows M=8..15 -> discard.
  if (half == 0) {
    #pragma unroll
    for (int r = 0; r < 8; ++r)
      out[r * DM + n0 + lm] = c[r] + bias[n0 + lm];
  }
}

// 24 blocks x 8 waves = 192 tiles = 3 matrices x 64 column-tiles.
__global__ __launch_bounds__(256) void qkv_proj_kernel(
    const float* __restrict__ q,
    const float* __restrict__ WQ, const float* __restrict__ bq,
    const float* __restrict__ WK, const float* __restrict__ bk,
    const float* __restrict__ WV, const float* __restrict__ bv,
    float* __restrict__ qh, float* __restrict__ knew, float* __restrict__ vnew)
{
  const int wave = threadIdx.x >> 5;
  const int lane = threadIdx.x & 31;
  const int tile = blockIdx.x * 8 + wave;   // 0..191 (uniform per wave)
  const int mat  = tile >> 6;               // 0=Q 1=K 2=V
  const int n0   = (tile & 63) << 4;

  const float* W = (mat == 0) ? WQ : (mat == 1) ? WK : WV;
  const float* b = (mat == 0) ? bq : (mat == 1) ? bk : bv;
  float*     dst = (mat == 0) ? qh : (mat == 1) ? knew : vnew;

  tile16_gemm_wmma(q, W, b, dst, n0, lane);
}

// 8 blocks x 8 waves = 64 column tiles of the output projection.
__global__ __launch_bounds__(256) void o_proj_kernel(
    const float* __restrict__ x,
    const float* __restrict__ WO, const float* __restrict__ bo,
    float* __restrict__ out)
{
  const int wave = threadIdx.x >> 5;
  const int lane = threadIdx.x & 31;
  const int n0   = (blockIdx.x * 8 + wave) << 4;
  tile16_gemm_wmma(x, WO, bo, out, n0, lane);
}

// cos/sin table: tab[t*32 + 2i] = cos(t * theta^{-i/16}), +1 = sin.
// t = 0..4096 (t=4096 doubles as the query position).
__global__ void rope_table_kernel(float* __restrict__ tab)
{
  const int idx = blockIdx.x * blockDim.x + threadIdx.x; // t*16 + i
  if (idx >= (CL + 1) * 16) return;
  const int t = idx >> 4;
  const int i = idx & 15;
  const float invf = powf(10000.0f, -(float)i / 16.0f);
  const float ang  = (float)t * invf;
  tab[t * 32 + 2 * i + 0] = cosf(ang);
  tab[t * 32 + 2 * i + 1] = sinf(ang);
}

// One block per (b,h). 128 threads; thread owns rows t, t+128, ... with an
// online-softmax register accumulator; single pass over k/v (256MB total).
__global__ __launch_bounds__(128) void attn_kernel(
    const float* __restrict__ qh,     // 8 x 1024 (projected, pre-RoPE)
    const float* __restrict__ knew,   // 8 x 1024
    const float* __restrict__ vnew,   // 8 x 1024
    const float* __restrict__ kcache, // 8 x 16 x 4096 x 64
    const float* __restrict__ vcache,
    const float* __restrict__ tab,    // (4097) x 32
    float* __restrict__ attn_out)     // 8 x 1024
{
  __shared__ float qs[HD];
  __shared__ float red[128];
  __shared__ float accs[128 * 65];    // stride-65 pad -> conflict-free

  const int tid = threadIdx.x;
  const int bh  = blockIdx.x;         // b*16 + h
  const int b   = bh >> 4;
  const int h   = bh & 15;

  const float* kc       = kcache + (size_t)bh * CL * HD;
  const float* vc       = vcache + (size_t)bh * CL * HD;
  const float* qrow     = qh   + b * DM + h * HD;
  const float* krow_new = knew + b * DM + h * HD;
  const float* vrow_new = vnew + b * DM + h * HD;

  // RoPE(q, t=4096) * 1/sqrt(64) into LDS (broadcast-read later).
  if (tid < HD) {
    float v = qrow[tid];
    float r;
    if (tid < 32) {
      const int i = tid >> 1;
      const float c = tab[CL * 32 + 2 * i + 0];
      const float s = tab[CL * 32 + 2 * i + 1];
      if ((tid & 1) == 0) r = v * c - qrow[tid + 1] * s;
      else                r = v * c + qrow[tid - 1] * s;
    } else {
      r = v;
    }
    qs[tid] = r * 0.125f;            // 1/sqrt(HD)
  }
  __syncthreads();

  float m_i = -INFINITY;
  float l_i = 0.0f;
  float acc[HD];
  #pragma unroll
  for (int d = 0; d < HD; ++d) acc[d] = 0.0f;

  for (int t = tid; t < CL + 1; t += 128) {
    const float* kr = (t < CL) ? (kc + (size_t)t * HD) : krow_new;
    const float* vr = (t < CL) ? (vc + (size_t)t * HD) : vrow_new;
    if (t + 128 < CL) {              // stream-ahead hint (global_prefetch_b8)
      __builtin_prefetch(kc + (size_t)(t + 128) * HD, 0, 1);
      __builtin_prefetch(vc + (size_t)(t + 128) * HD, 0, 1);
    }
    const float4* kr4 = (const float4*)kr;
    const float2* cs  = (const float2*)(tab + t * 32);

    float score = 0.0f;
    #pragma unroll
    for (int j = 0; j < 16; ++j) {
      float4 kk = kr4[j];
      if (j < 8) {                   // RoPE on dims 0..31 (pair-interleaved)
        const float2 c0 = cs[2 * j + 0];
        const float2 c1 = cs[2 * j + 1];
        const float x0 = kk.x, x1 = kk.y;
        kk.x = x0 * c0.x - x1 * c0.y;
        kk.y = x1 * c0.x + x0 * c0.y;
        const float y0 = kk.z, y1 = kk.w;
        kk.z = y0 * c1.x - y1 * c1.y;
        kk.w = y1 * c1.x + y0 * c1.y;
      }
      score += qs[4*j+0] * kk.x + qs[4*j+1] * kk.y
             + qs[4*j+2] * kk.z + qs[4*j+3] * kk.w;
    }

    const float newm      = fmaxf(m_i, score);
    const float scale_old = __expf(m_i - newm);   // exp(-inf)=0 on first iter
    const float p         = __expf(score - newm);
    l_i = l_i * scale_old + p;

    const float4* vr4 = (const float4*)vr;
    #pragma unroll
    for (int j = 0; j < 16; ++j) {
      const float4 vv = vr4[j];
      acc[4*j+0] = acc[4*j+0] * scale_old + p * vv.x;
      acc[4*j+1] = acc[4*j+1] * scale_old + p * vv.y;
      acc[4*j+2] = acc[4*j+2] * scale_old + p * vv.z;
      acc[4*j+3] = acc[4*j+3] * scale_old + p * vv.w;
    }
    m_i = newm;
  }

  // --- block combine: global max, rescale, sum ---
  red[tid] = m_i;
  __syncthreads();
  for (int off = 64; off > 0; off >>= 1) {
    if (tid < off) red[tid] = fmaxf(red[tid], red[tid + off]);
    __syncthreads();
  }
  const float M = red[0];
  __syncthreads();

  const float esc = __expf(m_i - M);
  red[tid] = l_i * esc;
  __syncthreads();
  for (int off = 64; off > 0; off >>= 1) {
    if (tid < off) red[tid] += red[tid + off];
    __syncthreads();
  }
  const float L = red[0];

  #pragma unroll
  for (int d = 0; d < HD; ++d) accs[tid * 65 + d] = acc[d] * esc;
  __syncthreads();

  if (tid < HD) {
    float s = 0.0f;
    for (int j = 0; j < 128; ++j) s += accs[j * 65 + tid];
    attn_out[b * DM + h * HD + tid] = s / L;
  }
}

extern "C" void kernel_launch(void* const* d_in, const int* in_sizes, int n_in,
                              void* d_out, int out_size, void* d_ws, size_t ws_size,
                              hipStream_t stream) {
  (void)in_sizes; (void)n_in; (void)out_size; (void)ws_size;
  const float* q      = (const float*)d_in[0];
  const float* kcache = (const float*)d_in[1];
  const float* vcache = (const float*)d_in[2];
  const float* WQ = (const float*)d_in[3];  const float* bq = (const float*)d_in[4];
  const float* WK = (const float*)d_in[5];  const float* bk = (const float*)d_in[6];
  const float* WV = (const float*)d_in[7];  const float* bv = (const float*)d_in[8];
  const float* WO = (const float*)d_in[9];  const float* bo = (const float*)d_in[10];
  float* out = (float*)d_out;

  float* ws   = (float*)d_ws;
  float* qh   = ws + 0;                 // 8192
  float* knew = ws + 8192;              // 8192
  float* vnew = ws + 16384;             // 8192
  float* attn = ws + 24576;             // 8192
  float* tab  = ws + 32768;             // 4097*32 = 131104

  qkv_proj_kernel<<<24, 256, 0, stream>>>(q, WQ, bq, WK, bk, WV, bv,
                                          qh, knew, vnew);
  rope_table_kernel<<<((CL + 1) * 16 + 255) / 256, 256, 0, stream>>>(tab);
  attn_kernel<<<BS_ * NH, 128, 0, stream>>>(qh, knew, vnew, kcache, vcache,
                                            tab, attn);
  o_proj_kernel<<<8, 256, 0, stream>>>(attn, WO, bo, out);
}